// SemiDefiniteLoss_25975962206274
// MI455X (gfx1250) — compile-verified
//
#include <hip/hip_runtime.h>

typedef float v4f __attribute__((ext_vector_type(4)));
typedef float v2f __attribute__((ext_vector_type(2)));
typedef float v8f __attribute__((ext_vector_type(8)));

#define N_ROWS 16384
#define N_COLS 2048
#define COL4   (N_COLS / 4)          // 512 float4-columns
#define BLK    256
#define ROWB   128                   // row-block count (grid.y)
#define ROWS_PER_BLK (N_ROWS / ROWB) // 128 rows per block

// ---------------------------------------------------------------------------
// Kernel 1: zero the 2048-float column-sum workspace (harness poisons d_ws).
// ---------------------------------------------------------------------------
__global__ void sdl_zero_ws(float* __restrict__ ws) {
    int i = blockIdx.x * blockDim.x + threadIdx.x;
    if (i < N_COLS) ws[i] = 0.0f;
}

// ---------------------------------------------------------------------------
// Kernel 2: bandwidth-bound streaming column sum-of-squares.
// Each thread owns 4 consecutive columns (one b128 load per row),
// accumulates ROWS_PER_BLK rows in registers, then 4 f32 global atomics.
// Non-temporal loads: data is streamed exactly once.
// ---------------------------------------------------------------------------
__global__ void sdl_colsumsq(const float* __restrict__ Y, float* __restrict__ ws) {
    const int col4 = blockIdx.x * BLK + threadIdx.x;      // 0..511
    const int row0 = blockIdx.y * ROWS_PER_BLK;

    const v4f* p = reinterpret_cast<const v4f*>(Y) + (size_t)row0 * COL4 + col4;

    v4f acc;
    acc.x = 0.0f; acc.y = 0.0f; acc.z = 0.0f; acc.w = 0.0f;

    #pragma unroll 8
    for (int r = 0; r < ROWS_PER_BLK; ++r) {
        v4f v = __builtin_nontemporal_load(p);
        acc.x = fmaf(v.x, v.x, acc.x);
        acc.y = fmaf(v.y, v.y, acc.y);
        acc.z = fmaf(v.z, v.z, acc.z);
        acc.w = fmaf(v.w, v.w, acc.w);
        p += COL4;
    }

    float* w = ws + col4 * 4;
    unsafeAtomicAdd(w + 0, acc.x);
    unsafeAtomicAdd(w + 1, acc.y);
    unsafeAtomicAdd(w + 2, acc.z);
    unsafeAtomicAdd(w + 3, acc.w);
}

// ---------------------------------------------------------------------------
// Kernel 3: finalize. per-column |1 - colsum/n|, then reduce 2048 -> 1.
// The 256 per-thread partials are reduced by wave 0 using chained
// V_WMMA_F32_16X16X4_F32 with A = all-ones: D[m][n] = sum_k B[k][n] (+C),
// so summing one D row over n yields the sum of ALL 64 B elements,
// independent of B's exact VGPR/lane layout. 4 chained WMMAs fold all 256
// partials; D's rows are identical, so the full-wave lane sum of c[0] is
// exactly 2x the answer.
// ---------------------------------------------------------------------------
__global__ void sdl_finalize(const float* __restrict__ ws, float* __restrict__ out) {
    __shared__ float part[256];

    const float inv_n = 1.0f / (float)N_ROWS;
    float s = 0.0f;
    for (int j = threadIdx.x; j < N_COLS; j += 256) {
        s += fabsf(1.0f - ws[j] * inv_n);
    }
    part[threadIdx.x] = s;
    __syncthreads();

    if (threadIdx.x < 32) {   // whole wave 0 active -> EXEC all ones (WMMA req.)
        v2f a; a.x = 1.0f; a.y = 1.0f;                 // A = ones(16x4)
        v8f c = {};                                     // C accumulator = 0
        #pragma unroll
        for (int chunk = 0; chunk < 4; ++chunk) {
            v2f b;
            b.x = part[chunk * 64 + threadIdx.x * 2 + 0];
            b.y = part[chunk * 64 + threadIdx.x * 2 + 1];
            c = __builtin_amdgcn_wmma_f32_16x16x4_f32(
                    /*neg_a=*/false, a, /*neg_b=*/false, b,
                    /*c_mod=*/(short)0, c, /*reuse_a=*/false, /*reuse_b=*/false);
        }
        // Sum c[0] (row 0 across lanes 0-15, row 8 across lanes 16-31;
        // all rows identical) over the full wave -> 2x total.
        float v = c[0];
        #pragma unroll
        for (int off = 16; off > 0; off >>= 1) {
            v += __shfl_down(v, off, 32);
        }
        if (threadIdx.x == 0) out[0] = v * 0.5f;
    }
}

// ---------------------------------------------------------------------------
extern "C" void kernel_launch(void* const* d_in, const int* in_sizes, int n_in,
                              void* d_out, int out_size, void* d_ws, size_t ws_size,
                              hipStream_t stream) {
    const float* Y   = (const float*)d_in[0];
    float*       out = (float*)d_out;
    float*       ws  = (float*)d_ws;   // needs N_COLS * 4 = 8 KB

    sdl_zero_ws <<<dim3((N_COLS + 255) / 256), dim3(256), 0, stream>>>(ws);
    sdl_colsumsq<<<dim3(COL4 / BLK, ROWB),     dim3(BLK), 0, stream>>>(Y, ws);
    sdl_finalize<<<dim3(1),                    dim3(256), 0, stream>>>(ws, out);
}